// Router_82738249990868
// MI455X (gfx1250) — compile-verified
//
#include <hip/hip_runtime.h>
#include <hip/hip_bf16.h>

typedef __attribute__((ext_vector_type(2))) float v2f;
typedef __attribute__((ext_vector_type(4))) float v4f;
typedef __attribute__((ext_vector_type(8))) float v8f;

#define N_TOK   8192
#define DIMK    4096
#define NEXP    16

// ---------------------------------------------------------------------------
// Repack gate (DIMK x 16 row-major) into a b128-per-lane WMMA B layout.
// K-octet ko owns 128 floats; lane l holds the 4 contiguous floats
//   gpack[ko*128 + l*4 + c] = gate[8*ko + 4*(l>>4) + c][l & 15],  c = 0..3
// WMMA#1 of the octet consumes (c0,c1)  -> K slots {8ko,8ko+1 | 8ko+4,8ko+5}
// WMMA#2 of the octet consumes (c2,c3)  -> K slots {8ko+2,8ko+3 | 8ko+6,8ko+7}
// (K-slot permutation is legal: WMMA sums symmetrically over its 4 K slots,
//  and the A-side loads use the exact same assignment.)
// ---------------------------------------------------------------------------
__global__ void gate_repack_kernel(const float* __restrict__ gate,
                                   float* __restrict__ gpack) {
    int o  = blockIdx.x * blockDim.x + threadIdx.x;   // [0, DIMK*NEXP)
    int ko = o >> 7;          // K-octet
    int r  = o & 127;
    int l  = r >> 2;          // lane 0..31
    int c  = r & 3;           // slot within lane's float4
    int h  = l >> 4;
    int n  = l & 15;          // expert column
    int k  = 8 * ko + 4 * h + c;
    gpack[o] = gate[k * NEXP + n];
}

// ---------------------------------------------------------------------------
// Phase A: f32 WMMA GEMM (8192x4096) x (4096x16), K split 2^KS_LOG2 ways.
// One wave per (16-token tile, K-split). Per K-octet: 1 A b128 + 1 B b128
// + 2 v_wmma_f32_16x16x4_f32. Compile-time trip count -> unroll 8 ->
// ~16 b128 loads in flight, pipelined s_wait_loadcnt schedule.
// ---------------------------------------------------------------------------
template <int KS_LOG2>
__global__ void __launch_bounds__(128)
router_gemm_kernel(const float* __restrict__ x,
                   const float* __restrict__ gpack,
                   float* __restrict__ part) {
    constexpr int KCHUNK = DIMK >> KS_LOG2;

    const int wave = (blockIdx.x * blockDim.x + threadIdx.x) >> 5;
    const int lane = threadIdx.x & 31;
    const int s    = wave & ((1 << KS_LOG2) - 1);   // K-split index
    const int tile = wave >> KS_LOG2;               // [0, 512)
    const int r    = lane & 15;                     // A row owned by lane
    const int h    = lane >> 4;                     // half-wave

    const float* __restrict__ xr = x + (size_t)(tile * 16 + r) * DIMK;

    v8f acc = {};
    const int k0 = s * KCHUNK;

#pragma unroll 8
    for (int k = k0; k < k0 + KCHUNK; k += 8) {
        // A: one b128 per lane covers this lane's 4 K slots for both WMMAs
        v4f a = *(const v4f*)(xr + k + 4 * h);
        // B: one b128 per lane from the pre-packed gate
        v4f b = *(const v4f*)(gpack + (size_t)(k >> 3) * 128 + lane * 4);

        v2f a0 = {a.x, a.y}, a1 = {a.z, a.w};
        v2f b0 = {b.x, b.y}, b1 = {b.z, b.w};

        acc = __builtin_amdgcn_wmma_f32_16x16x4_f32(false, a0, false, b0,
                                                    (short)0, acc, false, false);
        acc = __builtin_amdgcn_wmma_f32_16x16x4_f32(false, a1, false, b1,
                                                    (short)0, acc, false, false);
    }

    // C/D layout: VGPR j, lanes 0-15 -> token tile*16+j, lanes 16-31 -> +8.
    // part layout: [split][token][expert]  (64B contiguous per half-wave store)
    float* __restrict__ out = part + (size_t)s * N_TOK * NEXP
                                   + (size_t)(tile * 16 + 8 * h) * NEXP
                                   + (lane & 15);
#pragma unroll
    for (int j = 0; j < 8; ++j)
        out[j * NEXP] = acc[j];
}

// ---------------------------------------------------------------------------
// Phase B: one thread per token. Sum K-split partials (b128 loads), top-2 of
// 16 logits, sigmoid kept / 0 dropped, write (E,N) scores + (E,N) indices.
// ---------------------------------------------------------------------------
template <int KSPLIT>
__global__ void __launch_bounds__(256)
router_topk_kernel(const float* __restrict__ part, float* __restrict__ out) {
    const int t = blockIdx.x * blockDim.x + threadIdx.x;   // token [0, N_TOK)

    v4f v0 = {}, v1 = {}, v2 = {}, v3 = {};
#pragma unroll
    for (int s = 0; s < KSPLIT; ++s) {
        const v4f* __restrict__ p =
            (const v4f*)(part + (size_t)s * N_TOK * NEXP + (size_t)t * NEXP);
        v0 += p[0]; v1 += p[1]; v2 += p[2]; v3 += p[3];
    }

    float v[NEXP];
#pragma unroll
    for (int e = 0; e < 4; ++e) {
        v[e]      = v0[e];
        v[e + 4]  = v1[e];
        v[e + 8]  = v2[e];
        v[e + 12] = v3[e];
    }

    float m1 = -__builtin_inff(), m2 = -__builtin_inff();
#pragma unroll
    for (int e = 0; e < NEXP; ++e) {
        float val = v[e];
        if (val > m1)      { m2 = m1; m1 = val; }
        else if (val > m2) { m2 = val; }
    }

    float* __restrict__ idxout = out + (size_t)NEXP * N_TOK;
    const float tf = (float)t;
#pragma unroll
    for (int e = 0; e < NEXP; ++e) {
        float sc = (v[e] >= m2) ? (1.0f / (1.0f + __expf(-v[e]))) : 0.0f;
        out[(size_t)e * N_TOK + t]    = sc;
        idxout[(size_t)e * N_TOK + t] = tf;
    }
}

// ---------------------------------------------------------------------------
extern "C" void kernel_launch(void* const* d_in, const int* in_sizes, int n_in,
                              void* d_out, int out_size, void* d_ws, size_t ws_size,
                              hipStream_t stream) {
    const float* x    = (const float*)d_in[0];   // (8192, 4096) f32
    const float* gate = (const float*)d_in[1];   // (4096, 16)   f32
    float* out = (float*)d_out;                  // scores (16,8192) ++ indices (16,8192)

    float* gpack = (float*)d_ws;                 // DIMK*NEXP = 65536 floats (256 KB)
    float* part  = gpack + DIMK * NEXP;          // ksplit * N_TOK * NEXP floats

    // 1) repack gate into b128-per-lane WMMA B layout (65536 elements)
    gate_repack_kernel<<<(DIMK * NEXP) / 256, 256, 0, stream>>>(gate, gpack);

    // Pick K-split from available scratch (deterministic: ws_size is fixed).
    const size_t base = (size_t)DIMK * NEXP * sizeof(float);
    const size_t per_split = (size_t)N_TOK * NEXP * sizeof(float);   // 512 KB

    if (ws_size >= base + 8 * per_split) {
        // 4096 waves (512 tiles x 8 K-splits), 4 waves/block
        router_gemm_kernel<3><<<(512 << 3) / 4, 128, 0, stream>>>(x, gpack, part);
        router_topk_kernel<8><<<N_TOK / 256, 256, 0, stream>>>(part, out);
    } else if (ws_size >= base + 4 * per_split) {
        router_gemm_kernel<2><<<(512 << 2) / 4, 128, 0, stream>>>(x, gpack, part);
        router_topk_kernel<4><<<N_TOK / 256, 256, 0, stream>>>(part, out);
    } else {
        router_gemm_kernel<1><<<(512 << 1) / 4, 128, 0, stream>>>(x, gpack, part);
        router_topk_kernel<2><<<N_TOK / 256, 256, 0, stream>>>(part, out);
    }
}